// RHF_35708358099214
// MI455X (gfx1250) — compile-verified
//
#include <hip/hip_runtime.h>

typedef float v2f __attribute__((ext_vector_type(2)));
typedef float v8f __attribute__((ext_vector_type(8)));

#define SCF_ITERS 20
#define POWER_ITERS 24

// Per-workgroup: 128 threads = 4 waves, each wave owns 2 batches (8 batches/WG).
// LDS layout (floats):
//   MT   : 8 * (64*65)  = 33280   (M^T, row-padded to 65 for conflict-free access)
//   A_sh : 8 * 64 = 512
//   H_sh : 8 * 64 = 512
//   F_sh : 8 * 64 = 512
//   d_sh : 8 * 64 = 512   (vec(D) per batch)
//   tmp  : 4 * (16*17) = 1088  (per-wave 16x16 tile, padded to 17)
#define SMEM_FLOATS (8*4160 + 4*512 + 4*272)

__global__ __launch_bounds__(128, 1)
void rhf_scf_kernel(const float* __restrict__ Hg, const float* __restrict__ Ag,
                    const float* __restrict__ Gg, const float* __restrict__ Enucg,
                    float* __restrict__ out, int B)
{
    extern __shared__ float smem[];
    float* MT   = smem;                // 8*4160
    float* A_sh = MT + 8*4160;         // 8*64
    float* H_sh = A_sh + 512;
    float* F_sh = H_sh + 512;
    float* d_sh = F_sh + 512;
    float* tmp  = d_sh + 512;          // 4*272

    const int tid  = threadIdx.x;
    const int w    = tid >> 5;         // wave id 0..3
    const int lane = tid & 31;
    const int half = lane >> 4;        // 0 -> batch bl0, 1 -> batch bl1
    const int sub  = lane & 15;
    const int p    = sub & 7;          // row owned by this lane (duplicated on sub>=8)
    const int srcb = half << 4;        // shuffle source base
    const int bl0  = 2 * w, bl1 = bl0 + 1;
    const int bl   = bl0 + half;
    const int blockBase = blockIdx.x * 8;
    if (blockBase >= B) return;
    const int bb0 = blockBase + bl0, bb1 = blockBase + bl1;
    const int bbg = blockBase + bl;

    float* tmpw = tmp + w * 272;

    // ---------------- setup ----------------
    A_sh[bl0*64 + lane]      = Ag[(size_t)bb0*64 + lane];
    A_sh[bl0*64 + lane + 32] = Ag[(size_t)bb0*64 + lane + 32];
    A_sh[bl1*64 + lane]      = Ag[(size_t)bb1*64 + lane];
    A_sh[bl1*64 + lane + 32] = Ag[(size_t)bb1*64 + lane + 32];
    H_sh[bl0*64 + lane]      = Hg[(size_t)bb0*64 + lane];
    H_sh[bl0*64 + lane + 32] = Hg[(size_t)bb0*64 + lane + 32];
    H_sh[bl1*64 + lane]      = Hg[(size_t)bb1*64 + lane];
    H_sh[bl1*64 + lane + 32] = Hg[(size_t)bb1*64 + lane + 32];
    d_sh[bl0*64 + lane] = 0.0f;  d_sh[bl0*64 + lane + 32] = 0.0f;
    d_sh[bl1*64 + lane] = 0.0f;  d_sh[bl1*64 + lane + 32] = 0.0f;

    // M^T = (2*G[pqrs] - G[prqs])^T, stored as MT[rs*65 + pq] (conflict-free both ways)
    for (int bi = 0; bi < 2; ++bi) {
        const int blc = bl0 + bi;
        const float* Gb = Gg + (size_t)(blockBase + blc) * 4096;
        float* MTb = MT + blc * 4160;
        __builtin_prefetch(Gb + lane * 16, 0, 0);   // global_prefetch_b8 on the G stream
        #pragma unroll 4
        for (int i = 0; i < 128; ++i) {
            const int t  = i * 32 + lane;           // == pq*64 + rs == G linear index
            const int pq = t >> 6, rs = t & 63;
            const int pp = pq >> 3, qq = pq & 7, rr = rs >> 3, ss = rs & 7;
            const float g1 = Gb[t];                                   // G[p,q,r,s] (coalesced)
            const float g2 = Gb[((pp*8 + rr)*8 + qq)*8 + ss];          // G[p,r,q,s] (L0/L2 hit)
            MTb[rs*65 + pq] = 2.0f * g1 - g2;
        }
    }
    __syncthreads();

    // Per-lane row of A for c = A x
    float Arow[8];
    #pragma unroll
    for (int q = 0; q < 8; ++q) Arow[q] = A_sh[bl*64 + p*8 + q];

    // Prebuild block-diagonal A operand fragments (constant across all iterations).
    // 16x16x4 f32 WMMA, K=16 split into 4 slices kc. A-op: lane holds row m=sub,
    // k = 4*kc + 2*half + j. B-op: lane holds col n=sub, same k mapping.
    v2f afrag[4], abfrag[4];
    #pragma unroll
    for (int kc = 0; kc < 4; ++kc) {
        const int kb = 4*kc + 2*half;
        #pragma unroll
        for (int j = 0; j < 2; ++j) {
            const int k = kb + j;
            const float va = A_sh[(bl0 + (k >> 3))*64 + (sub & 7)*8 + (k & 7)]; // Ablk[m=sub][k]
            afrag[kc][j]  = ((k >> 3) == (sub >> 3)) ? va : 0.0f;
            const float vb = A_sh[(bl0 + (k >> 3))*64 + (k & 7)*8 + (sub & 7)]; // Ablk[k][n=sub]
            abfrag[kc][j] = ((k >> 3) == (sub >> 3)) ? vb : 0.0f;
        }
    }

    float xv = 0.35355339f;   // warm-started power-iteration vector (1/sqrt(8))
    float cv = 0.0f;

    for (int iter = 0; iter < SCF_ITERS; ++iter) {
        // ---- Phase A: F = H + M * vec(D)  (4 rows per lane, conflict-free LDS) ----
        float a00 = 0.f, a01 = 0.f, a10 = 0.f, a11 = 0.f;
        const float* MT0 = MT + bl0 * 4160;
        const float* MT1 = MT + bl1 * 4160;
        const float* dp0 = d_sh + bl0 * 64;
        const float* dp1 = d_sh + bl1 * 64;
        #pragma unroll 8
        for (int k = 0; k < 64; ++k) {
            const float dk0 = dp0[k], dk1 = dp1[k];   // LDS broadcast
            a00 = fmaf(MT0[k*65 + lane],      dk0, a00);
            a01 = fmaf(MT0[k*65 + lane + 32], dk0, a01);
            a10 = fmaf(MT1[k*65 + lane],      dk1, a10);
            a11 = fmaf(MT1[k*65 + lane + 32], dk1, a11);
        }
        F_sh[bl0*64 + lane]      = H_sh[bl0*64 + lane]      + a00;
        F_sh[bl0*64 + lane + 32] = H_sh[bl0*64 + lane + 32] + a01;
        F_sh[bl1*64 + lane]      = H_sh[bl1*64 + lane]      + a10;
        F_sh[bl1*64 + lane + 32] = H_sh[bl1*64 + lane + 32] + a11;
        __builtin_amdgcn_wave_barrier();

        // ---- Phase B: Fp = Ablk * Fblk * Ablk via f32 WMMA (block-diagonal 2 batches) ----
        v8f tacc = {0.f,0.f,0.f,0.f,0.f,0.f,0.f,0.f};
        #pragma unroll
        for (int kc = 0; kc < 4; ++kc) {
            const int kb = 4*kc + 2*half;
            v2f bf;
            #pragma unroll
            for (int j = 0; j < 2; ++j) {
                const int k = kb + j;
                const float v = F_sh[(bl0 + (k >> 3))*64 + (k & 7)*8 + (sub & 7)]; // Fblk[k][n]
                bf[j] = ((k >> 3) == (sub >> 3)) ? v : 0.0f;
            }
            tacc = __builtin_amdgcn_wmma_f32_16x16x4_f32(
                false, afrag[kc], false, bf, (short)0, tacc, false, false);
        }
        #pragma unroll
        for (int v = 0; v < 8; ++v) tmpw[(v + 8*half)*17 + sub] = tacc[v];  // C-layout -> tile
        __builtin_amdgcn_wave_barrier();

        v8f fpacc = {0.f,0.f,0.f,0.f,0.f,0.f,0.f,0.f};
        #pragma unroll
        for (int kc = 0; kc < 4; ++kc) {
            const int kb = 4*kc + 2*half;
            v2f ta;
            ta[0] = tmpw[sub*17 + kb];        // T as A-operand: row m=sub, k slice
            ta[1] = tmpw[sub*17 + kb + 1];
            fpacc = __builtin_amdgcn_wmma_f32_16x16x4_f32(
                false, ta, false, abfrag[kc], (short)0, fpacc, false, false);
        }
        #pragma unroll
        for (int v = 0; v < 8; ++v) tmpw[(v + 8*half)*17 + sub] = fpacc[v]; // Fp tile
        __builtin_amdgcn_wave_barrier();

        // ---- Phase C: lowest eigenvector of Fp (8x8) via shifted power iteration ----
        float fprow[8];
        #pragma unroll
        for (int q = 0; q < 8; ++q) fprow[q] = tmpw[(8*half + p)*17 + 8*half + q];
        float asum = 0.f;
        #pragma unroll
        for (int q = 0; q < 8; ++q) asum += fabsf(fprow[q]);
        float g = fprow[p] + asum - fabsf(fprow[p]);          // Gershgorin row bound
        float sigma = g;
        #pragma unroll
        for (int q = 0; q < 8; ++q) sigma = fmaxf(sigma, __shfl(g, srcb + q));
        sigma += 1e-3f;
        float Srow[8];
        #pragma unroll
        for (int q = 0; q < 8; ++q) Srow[q] = ((q == p) ? sigma : 0.0f) - fprow[q];

        #pragma unroll 4
        for (int it = 0; it < POWER_ITERS; ++it) {
            float y = 0.f;
            #pragma unroll
            for (int q = 0; q < 8; ++q) y = fmaf(Srow[q], __shfl(xv, srcb + q), y);
            float nn = 1e-30f;
            #pragma unroll
            for (int q = 0; q < 8; ++q) { const float yq = __shfl(y, srcb + q); nn = fmaf(yq, yq, nn); }
            xv = y * rsqrtf(nn);
        }

        cv = 0.f;                                            // c = A * v_min
        #pragma unroll
        for (int q = 0; q < 8; ++q) cv = fmaf(Arow[q], __shfl(xv, srcb + q), cv);
        #pragma unroll
        for (int q = 0; q < 8; ++q)                          // D = c c^T (dup lanes write same val)
            d_sh[bl*64 + p*8 + q] = cv * __shfl(cv, srcb + q);
        __builtin_amdgcn_wave_barrier();
    }

    // ---- Energy: E = sum((F+H) .* D) + Enuc = c^T (F+H) c + Enuc ----
    float s = 0.f;
    #pragma unroll
    for (int q = 0; q < 8; ++q)
        s = fmaf(F_sh[bl*64 + p*8 + q] + H_sh[bl*64 + p*8 + q], __shfl(cv, srcb + q), s);
    s *= cv;
    float E = 0.f;
    #pragma unroll
    for (int q = 0; q < 8; ++q) E += __shfl(s, srcb + q);
    if (sub == 0) out[bbg] = E + Enucg[bbg];
}

extern "C" void kernel_launch(void* const* d_in, const int* in_sizes, int n_in,
                              void* d_out, int out_size, void* d_ws, size_t ws_size,
                              hipStream_t stream) {
    const float* H    = (const float*)d_in[0];
    const float* A    = (const float*)d_in[1];
    const float* G    = (const float*)d_in[2];
    const float* Enuc = (const float*)d_in[3];
    float* out = (float*)d_out;
    const int B = in_sizes[3];              // batch count (Enuc elements)
    const int blocks = (B + 7) / 8;
    const size_t smem_bytes = (size_t)SMEM_FLOATS * sizeof(float);  // 145,664 B
    rhf_scf_kernel<<<blocks, 128, smem_bytes, stream>>>(H, A, G, Enuc, out, B);
}